// AdditiveAttention_65987877535812
// MI455X (gfx1250) — compile-verified
//
#include <hip/hip_runtime.h>
#include <stdint.h>

typedef float v2f __attribute__((ext_vector_type(2)));
typedef float v8f __attribute__((ext_vector_type(8)));
typedef int   v2i __attribute__((ext_vector_type(2)));

#define B_SZ   1024
#define A_SZ   512
#define DV_SZ  2048
#define DT_SZ  512
#define OUT_STRIDE (DV_SZ + DT_SZ)

__device__ __forceinline__ float fast_tanh(float x) {
#if __has_builtin(__builtin_amdgcn_tanhf)
    return __builtin_amdgcn_tanhf(x);
#else
    float r;
    asm volatile("v_tanh_f32 %0, %1" : "=v"(r) : "v"(x));
    return r;
#endif
}

// P = X @ W^T + bias   (X: (M,K) row-major, W: (N,K) row-major, P: (M,N))
// One wave computes one 16x16 output tile with V_WMMA_F32_16X16X4_F32.
// Manual 1-stage software pipeline: loads for k+8 issue before the WMMAs
// consuming k, so s_wait_loadcnt covers loads a full WMMA-pair old.
__global__ __launch_bounds__(128) void wmma_linear_f32(
    const float* __restrict__ X,
    const float* __restrict__ W,
    const float* __restrict__ bias,
    float* __restrict__ P,
    int K, int N)
{
    const int lane = threadIdx.x & 31;
    const int wave = threadIdx.x >> 5;
    const int n0   = blockIdx.x * 16;
    const int m0   = (blockIdx.y * 4 + wave) * 16;
    const int half = lane >> 4;     // 0: K=0,1 ; 1: K=2,3 (f32 A/B VGPR layout)
    const int l    = lane & 15;

    // A 16x4 f32: lane l holds row M=m0+l, cols {k+2*half, k+2*half+1}
    const float* pa = X + (size_t)(m0 + l) * K + 2 * half;
    // B 4x16 f32: lane l holds col N=n0+l, rows {k+2*half, k+2*half+1} of W^T
    const float* pb = W + (size_t)(n0 + l) * K + 2 * half;

    // C/D layout: VGPR r -> M = m0 + r + 8*half, N = n0 + l. Bias depends on N only.
    const float bval = bias[n0 + l];
    v8f acc0, acc1;
#pragma unroll
    for (int r = 0; r < 8; ++r) { acc0[r] = bval; acc1[r] = 0.0f; }

    v2f a0 = *(const v2f*)(pa);
    v2f a1 = *(const v2f*)(pa + 4);
    v2f b0 = *(const v2f*)(pb);
    v2f b1 = *(const v2f*)(pb + 4);

    int k = 0;
    for (; k < K - 8; k += 8) {
        // one prefetch per 128B cacheline, 1KB ahead
        if (((k + 8) & 31) == 0) {
            __builtin_prefetch(pa + k + 256, 0, 3);
            __builtin_prefetch(pb + k + 256, 0, 3);
        }
        v2f na0 = *(const v2f*)(pa + k + 8);
        v2f na1 = *(const v2f*)(pa + k + 12);
        v2f nb0 = *(const v2f*)(pb + k + 8);
        v2f nb1 = *(const v2f*)(pb + k + 12);
        acc0 = __builtin_amdgcn_wmma_f32_16x16x4_f32(
                   false, a0, false, b0, (short)0, acc0, false, false);
        acc1 = __builtin_amdgcn_wmma_f32_16x16x4_f32(
                   false, a1, false, b1, (short)0, acc1, false, false);
        a0 = na0; a1 = na1; b0 = nb0; b1 = nb1;
    }
    acc0 = __builtin_amdgcn_wmma_f32_16x16x4_f32(
               false, a0, false, b0, (short)0, acc0, false, false);
    acc1 = __builtin_amdgcn_wmma_f32_16x16x4_f32(
               false, a1, false, b1, (short)0, acc1, false, false);

#pragma unroll
    for (int r = 0; r < 8; ++r) {
        P[(size_t)(m0 + 8 * half + r) * N + n0 + l] = acc0[r] + acc1[r];
    }
}

// One block per batch row. Fuses: scores (tanh reduction), softmax,
// attended = ts * softmax, and concat [vision | attended] into out.
// Never materializes the (B, A, A) intermediate. LDS staging of vp / v_w
// uses GLOBAL_LOAD_ASYNC_TO_LDS_B64 (+ s_wait_asynccnt) when available.
__global__ __launch_bounds__(256) void additive_attn(
    const float* __restrict__ vp,   // (B, A)
    const float* __restrict__ tp,   // (B, A)
    const float* __restrict__ ts,   // (B, DT)
    const float* __restrict__ vis,  // (B, DV)
    const float* __restrict__ v_w,  // (A,)
    const float* __restrict__ v_b,  // (1,)
    float* __restrict__ out)        // (B, DV+DT)
{
    __shared__ float svp[A_SZ];
    __shared__ float svw[A_SZ];
    __shared__ float red[256];

    const int b = blockIdx.x;
    const int t = threadIdx.x;

    const float* vprow = vp + (size_t)b * A_SZ;
    const float* tprow = tp + (size_t)b * A_SZ;

#if __has_builtin(__builtin_amdgcn_global_load_async_to_lds_b64)
    // each thread moves 8B of vp-row and 8B of v_w into LDS asynchronously
    __builtin_amdgcn_global_load_async_to_lds_b64(
        (v2i*)(vprow + 2 * t), (v2i*)(&svp[2 * t]), 0, 0);
    __builtin_amdgcn_global_load_async_to_lds_b64(
        (v2i*)(v_w + 2 * t), (v2i*)(&svw[2 * t]), 0, 0);
#if __has_builtin(__builtin_amdgcn_s_wait_asynccnt)
    __builtin_amdgcn_s_wait_asynccnt(0);
#else
    asm volatile("s_wait_asynccnt 0x0" ::: "memory");
#endif
#else
    svp[t]       = vprow[t];
    svp[t + 256] = vprow[t + 256];
    svw[t]       = v_w[t];
    svw[t + 256] = v_w[t + 256];
#endif
    __syncthreads();

    const float vb = v_b[0];
    float sc[2];
#pragma unroll
    for (int h = 0; h < 2; ++h) {
        const int   i   = t + h * 256;
        const float tpi = tprow[i];
        float s = vb;
#pragma unroll 8
        for (int j = 0; j < A_SZ; ++j) {
            s += svw[j] * fast_tanh(tpi + svp[j]);   // j uniform -> LDS broadcast
        }
        sc[h] = s;
    }

    // softmax max-reduce over 512 scores
    red[t] = fmaxf(sc[0], sc[1]);
    __syncthreads();
    for (int off = 128; off > 0; off >>= 1) {
        if (t < off) red[t] = fmaxf(red[t], red[t + off]);
        __syncthreads();
    }
    const float m = red[0];
    __syncthreads();

    const float e0 = __expf(sc[0] - m);
    const float e1 = __expf(sc[1] - m);
    red[t] = e0 + e1;
    __syncthreads();
    for (int off = 128; off > 0; off >>= 1) {
        if (t < off) red[t] += red[t + off];
        __syncthreads();
    }
    const float inv = 1.0f / red[0];

    // attended_ts into the concat tail
    const float* tsrow = ts + (size_t)b * DT_SZ;
    float*       orow  = out + (size_t)b * OUT_STRIDE;
    orow[DV_SZ + t]       = tsrow[t]       * e0 * inv;
    orow[DV_SZ + t + 256] = tsrow[t + 256] * e1 * inv;

    // vision copy (head of concat), vectorized float4: 512 float4 / row
    const float4* vsrc = (const float4*)(vis + (size_t)b * DV_SZ);
    float4*       vdst = (float4*)orow;
    vdst[t]       = vsrc[t];
    vdst[t + 256] = vsrc[t + 256];
}

extern "C" void kernel_launch(void* const* d_in, const int* in_sizes, int n_in,
                              void* d_out, int out_size, void* d_ws, size_t ws_size,
                              hipStream_t stream) {
    const float* vision = (const float*)d_in[0];   // (1024, 2048)
    const float* ts     = (const float*)d_in[1];   // (1024, 512)
    const float* Wv_w   = (const float*)d_in[2];   // (512, 2048)
    const float* Wv_b   = (const float*)d_in[3];   // (512,)
    const float* Wt_w   = (const float*)d_in[4];   // (512, 512)
    const float* Wt_b   = (const float*)d_in[5];   // (512,)
    const float* v_w    = (const float*)d_in[6];   // (512,)
    const float* v_b    = (const float*)d_in[7];   // (1,)
    float* out = (float*)d_out;                    // (1024, 2560)

    float* vp = (float*)d_ws;                      // (1024, 512)
    float* tp = vp + (size_t)B_SZ * A_SZ;          // (1024, 512)

    // vp = vision @ Wv^T + Wv_b   : grid (512/16, 1024/64), 4 waves/block
    wmma_linear_f32<<<dim3(A_SZ / 16, B_SZ / 64), 128, 0, stream>>>(
        vision, Wv_w, Wv_b, vp, DV_SZ, A_SZ);
    // tp = ts @ Wt^T + Wt_b
    wmma_linear_f32<<<dim3(A_SZ / 16, B_SZ / 64), 128, 0, stream>>>(
        ts, Wt_w, Wt_b, tp, DT_SZ, A_SZ);
    // fused tanh-score / softmax / scale / concat
    additive_attn<<<B_SZ, 256, 0, stream>>>(vp, tp, ts, vision, v_w, v_b, out);
}